// GraphEncoder_62869731279388
// MI455X (gfx1250) — compile-verified
//
#include <hip/hip_runtime.h>
#include <math.h>

// ---- problem constants (match reference) ----
#define DM 128          // d_model
#define NH 8            // heads
#define DH 16           // d_head
#define EDIM 64         // edge_dim
#define NEG_SLOPE 0.01f
#define EPS_BN 1e-5f

typedef __attribute__((ext_vector_type(2))) float v2f;
typedef __attribute__((ext_vector_type(8))) float v8f;

// ---------------------------------------------------------------------------
// sign-aware float atomic max (amax initialized to -inf)
// ---------------------------------------------------------------------------
__device__ __forceinline__ void atomicMaxF(float* addr, float val) {
    if (val >= 0.0f)
        atomicMax((int*)addr, __float_as_int(val));
    else
        atomicMin((unsigned int*)addr, (unsigned int)__float_as_int(val));
}

// ---------------------------------------------------------------------------
// Generic WMMA GEMM:  out[M x 128] = A[M x Kdim] @ W[Kdim x 128] + bias[128]
// 16-row tile per block, 8 waves, wave w -> cols [16w,16w+16).
// A tile is streamed global->LDS with GLOBAL_LOAD_ASYNC_TO_LDS_B128
// (ASYNCcnt), then consumed by V_WMMA_F32_16X16X4_F32 with a K-step of 4.
// A-frag (16x4 f32): lanes 0-15: M=lane, v0=K0,v1=K1; lanes 16-31: v0=K2,v1=K3
// C/D (16x16 f32):   VGPR r: lanes 0-15 -> M=r, lanes 16-31 -> M=r+8, N=lane&15
// ---------------------------------------------------------------------------
__global__ __launch_bounds__(256) void wmma_gemm_bias(
    const float* __restrict__ A, const float* __restrict__ W,
    const float* __restrict__ bias, float* __restrict__ out,
    int M, int Kdim)
{
    __shared__ __align__(16) float at[16][132];   // padded: stride 132 breaks bank conflicts
    const int tid  = threadIdx.x;
    const int row0 = blockIdx.x * 16;

    // async global -> LDS copy of the 16 x Kdim A-tile (float4 granularity)
    const int kq  = Kdim >> 2;                    // float4s per row (16 or 32)
    const int nf4 = 16 * kq;
    for (int i = tid; i < nf4; i += 256) {
        int r  = i / kq;
        int c4 = i - r * kq;
        int row = row0 + r;
        if (row >= M) row = M - 1;                // clamp; OOB rows never stored
        unsigned long long ga =
            (unsigned long long)(uintptr_t)(A + (size_t)row * Kdim + ((size_t)c4 << 2));
        unsigned lo = (unsigned)(uintptr_t)(&at[r][0]) + ((unsigned)c4 << 4);
        asm volatile("global_load_async_to_lds_b128 %0, %1, off"
                     :: "v"(lo), "v"(ga) : "memory");
    }
    asm volatile("s_wait_asynccnt 0x0" ::: "memory");
    __syncthreads();

    const int lane  = tid & 31;
    const int wave  = tid >> 5;                   // 0..7 -> output col tile
    const int m     = lane & 15;
    const int khalf = (lane >> 4) << 1;           // 0 (lanes 0-15) or 2 (lanes 16-31)
    const int n0    = wave * 16;

    // pull this wave's weight panel toward the caches (global_prefetch_b8)
    __builtin_prefetch(W + n0 + m, 0, 3);
    __builtin_prefetch(W + (size_t)(Kdim >> 1) * DM + n0 + m, 0, 3);

    const float bv = bias[n0 + m];                // bias depends only on column
    v8f acc;
    #pragma unroll
    for (int i = 0; i < 8; ++i) acc[i] = bv;

    for (int k0 = 0; k0 < Kdim; k0 += 4) {
        const int kb = k0 + khalf;
        v2f a, b;
        a.x = at[m][kb];
        a.y = at[m][kb + 1];
        b.x = W[(size_t)kb       * DM + n0 + m];
        b.y = W[(size_t)(kb + 1) * DM + n0 + m];
        acc = __builtin_amdgcn_wmma_f32_16x16x4_f32(
                  false, a, false, b, (short)0, acc, false, false);
    }

    const int rofs = (lane >> 4) << 3;            // 0 or 8
    #pragma unroll
    for (int r = 0; r < 8; ++r) {
        int row = row0 + r + rofs;
        if (row < M) out[(size_t)row * DM + n0 + m] = acc[r];
    }
}

// ---------------------------------------------------------------------------
// per-layer init: amax = -inf, denom = 0, bn stats = 0
// ---------------------------------------------------------------------------
__global__ void init_kernel(float* amax, float* denom, float* sums, int nh)
{
    int t = blockIdx.x * blockDim.x + threadIdx.x;
    if (t < nh) { amax[t] = -__builtin_inff(); denom[t] = 0.0f; }
    if (t < 2 * DM) sums[t] = 0.0f;
}

// ---------------------------------------------------------------------------
// alpha[e,h] = dot(q[dst[e],h,:], k[src[e],h,:] + e[e,h,:]) / sqrt(16)
// + segment max into amax[dst,h].  One thread per (edge, head).
// ---------------------------------------------------------------------------
__global__ __launch_bounds__(256) void alpha_max_kernel(
    const float* __restrict__ q, const float* __restrict__ k,
    const float* __restrict__ ef, const int* __restrict__ src,
    const int* __restrict__ dst, float* __restrict__ alpha,
    float* __restrict__ amax, int E)
{
    int t = blockIdx.x * blockDim.x + threadIdx.x;
    if (t >= E * NH) return;
    int ed = t >> 3, h = t & 7;
    int s = src[ed], d = dst[ed];
    const float4* qp = (const float4*)(q  + (size_t)d  * DM + h * DH);
    const float4* kp = (const float4*)(k  + (size_t)s  * DM + h * DH);
    const float4* ep = (const float4*)(ef + (size_t)ed * DM + h * DH);
    float acc = 0.f;
    #pragma unroll
    for (int i = 0; i < 4; ++i) {
        float4 qv = qp[i], kv = kp[i], ev = ep[i];
        acc += qv.x * (kv.x + ev.x) + qv.y * (kv.y + ev.y)
             + qv.z * (kv.z + ev.z) + qv.w * (kv.w + ev.w);
    }
    acc *= 0.25f;                                 // 1/sqrt(d_head)
    alpha[t] = acc;
    atomicMaxF(&amax[d * NH + h], acc);
}

// ---------------------------------------------------------------------------
// alpha <- exp(alpha - amax[dst]);  denom[dst] += alpha
// ---------------------------------------------------------------------------
__global__ __launch_bounds__(256) void exp_sum_kernel(
    float* __restrict__ alpha, const float* __restrict__ amax,
    float* __restrict__ denom, const int* __restrict__ dst, int E)
{
    int t = blockIdx.x * blockDim.x + threadIdx.x;
    if (t >= E * NH) return;
    int ed = t >> 3, h = t & 7;
    int d = dst[ed];
    float ex = __expf(alpha[t] - amax[d * NH + h]);
    alpha[t] = ex;
    atomicAdd(&denom[d * NH + h], ex);
}

// ---------------------------------------------------------------------------
// out[dst,h,:] += (v[src,h,:] + e[e,h,:]) * alpha / (denom[dst,h] + 1e-16)
// out was pre-initialized with x@Ws + bs (root weight) by the skip GEMM.
// ---------------------------------------------------------------------------
__global__ __launch_bounds__(256) void message_kernel(
    const float* __restrict__ vv, const float* __restrict__ ef,
    const float* __restrict__ alpha, const float* __restrict__ denom,
    const int* __restrict__ src, const int* __restrict__ dst,
    float* __restrict__ out, int E)
{
    int t = blockIdx.x * blockDim.x + threadIdx.x;
    if (t >= E * NH) return;
    int ed = t >> 3, h = t & 7;
    int s = src[ed], d = dst[ed];
    float w = alpha[t] / (denom[d * NH + h] + 1e-16f);
    const float4* vp = (const float4*)(vv + (size_t)s  * DM + h * DH);
    const float4* ep = (const float4*)(ef + (size_t)ed * DM + h * DH);
    float* op = out + (size_t)d * DM + h * DH;
    #pragma unroll
    for (int i = 0; i < 4; ++i) {
        float4 va = vp[i], ea = ep[i];
        atomicAdd(op + 4 * i + 0, (va.x + ea.x) * w);
        atomicAdd(op + 4 * i + 1, (va.y + ea.y) * w);
        atomicAdd(op + 4 * i + 2, (va.z + ea.z) * w);
        atomicAdd(op + 4 * i + 3, (va.w + ea.w) * w);
    }
}

// ---------------------------------------------------------------------------
// BatchNorm statistics: per-channel sum and sum-of-squares (atomics at end)
// ---------------------------------------------------------------------------
__global__ __launch_bounds__(128) void bn_stats_kernel(
    const float* __restrict__ out, float* __restrict__ sums, int N)
{
    int c = threadIdx.x;                          // channel 0..127
    float s = 0.f, s2 = 0.f;
    for (int r = blockIdx.x; r < N; r += gridDim.x) {
        float v = out[(size_t)r * DM + c];
        s += v; s2 += v * v;
    }
    atomicAdd(&sums[c], s);
    atomicAdd(&sums[DM + c], s2);
}

// ---------------------------------------------------------------------------
// BN apply (training stats, biased var) + LeakyReLU -> next-layer x
// ---------------------------------------------------------------------------
__global__ __launch_bounds__(256) void bn_apply_kernel(
    const float* __restrict__ out, const float* __restrict__ sums,
    const float* __restrict__ gamma, const float* __restrict__ beta,
    float* __restrict__ xnext, int N)
{
    int t = blockIdx.x * blockDim.x + threadIdx.x;
    if (t >= N * DM) return;
    int c = t & (DM - 1);
    float invN = 1.0f / (float)N;
    float mu  = sums[c] * invN;
    float var = sums[DM + c] * invN - mu * mu;
    float y = (out[t] - mu) * rsqrtf(var + EPS_BN) * gamma[c] + beta[c];
    xnext[t] = (y >= 0.f) ? y : NEG_SLOPE * y;
}

// ---------------------------------------------------------------------------
extern "C" void kernel_launch(void* const* d_in, const int* in_sizes, int n_in,
                              void* d_out, int out_size, void* d_ws, size_t ws_size,
                              hipStream_t stream)
{
    const float* x_in  = (const float*)d_in[0];
    const int*   eidx  = (const int*)  d_in[1];
    const float* eattr = (const float*)d_in[2];
    const float* Wq = (const float*)d_in[3];
    const float* bq = (const float*)d_in[4];
    const float* Wk = (const float*)d_in[5];
    const float* bk = (const float*)d_in[6];
    const float* Wv = (const float*)d_in[7];
    const float* bv = (const float*)d_in[8];
    const float* We = (const float*)d_in[9];
    const float* be = (const float*)d_in[10];
    const float* Wsk = (const float*)d_in[11];
    const float* bsk = (const float*)d_in[12];
    const float* gamma = (const float*)d_in[13];
    const float* beta  = (const float*)d_in[14];

    const int N = in_sizes[0] / DM;
    const int E = in_sizes[1] / 2;
    const int L = in_sizes[4] / DM;               // bq is [L, D]

    const int* src = eidx;
    const int* dst = eidx + E;

    // workspace layout (floats)
    float* ws = (float*)d_ws;
    const size_t ND = (size_t)N * DM;
    float* qb    = ws;
    float* kb    = qb + ND;
    float* vb    = kb + ND;
    float* ob    = vb + ND;            // skip + aggregated messages
    float* xb    = ob + ND;            // x for next layer
    float* alpha = xb + ND;            // [E, H]
    float* amax  = alpha + (size_t)E * NH;
    float* denom = amax + (size_t)N * NH;
    float* sums  = denom + (size_t)N * NH;        // [2, 128]
    float* eb    = sums + 2 * DM;      // [E, 128]  (largest, last)

    const int ntileN = (N + 15) / 16;
    const int ntileE = (E + 15) / 16;
    const int gEH    = (E * NH + 255) / 256;
    const int gInit  = (N * NH + 255) / 256;
    const int gNE    = (int)((ND + 255) / 256);

    for (int l = 0; l < L; ++l) {
        const float* xcur  = (l == 0) ? x_in : xb;
        float*       xnext = (l == L - 1) ? (float*)d_out : xb;
        const size_t oD  = (size_t)l * DM * DM;
        const size_t oE  = (size_t)l * EDIM * DM;
        const size_t oB  = (size_t)l * DM;

        init_kernel<<<gInit, 256, 0, stream>>>(amax, denom, sums, N * NH);

        wmma_gemm_bias<<<ntileN, 256, 0, stream>>>(xcur, Wq + oD, bq + oB, qb, N, DM);
        wmma_gemm_bias<<<ntileN, 256, 0, stream>>>(xcur, Wk + oD, bk + oB, kb, N, DM);
        wmma_gemm_bias<<<ntileN, 256, 0, stream>>>(xcur, Wv + oD, bv + oB, vb, N, DM);
        wmma_gemm_bias<<<ntileN, 256, 0, stream>>>(xcur, Wsk + oD, bsk + oB, ob, N, DM);
        wmma_gemm_bias<<<ntileE, 256, 0, stream>>>(eattr, We + oE, be + oB, eb, E, EDIM);

        alpha_max_kernel<<<gEH, 256, 0, stream>>>(qb, kb, eb, src, dst, alpha, amax, E);
        exp_sum_kernel  <<<gEH, 256, 0, stream>>>(alpha, amax, denom, dst, E);
        message_kernel  <<<gEH, 256, 0, stream>>>(vb, eb, alpha, denom, src, dst, ob, E);

        bn_stats_kernel<<<256, 128, 0, stream>>>(ob, sums, N);
        bn_apply_kernel<<<gNE, 256, 0, stream>>>(ob, sums, gamma + oB, beta + oB, xnext, N);
    }
}